// WinGNN_52132313039370
// MI455X (gfx1250) — compile-verified
//
#include <hip/hip_runtime.h>
#include <hip/hip_bf16.h>

// WinGNN forward for MI455X (gfx1250, wave32).
// - GEMMs: V_WMMA_F32_16X16X4_F32, B-matrix staged in LDS via
//   global_load_async_to_lds_b128 (padded rows -> conflict-free ds_load_b64).
// - Propagation: float4 gather + hardware f32 scatter atomics (L2-resident).

typedef __attribute__((ext_vector_type(2))) float v2f;
typedef __attribute__((ext_vector_type(8))) float v8f;

#define DIM 128
#define PAD1 130   // LDS row stride (floats) for 128-wide W1
#define PAD2 258   // LDS row stride (floats) for 256-wide W_l1

// ---- gfx1250 async global->LDS copy (16B per lane) ----
__device__ __forceinline__ void async_lds_b128(unsigned lds_addr, const void* gptr) {
    asm volatile("global_load_async_to_lds_b128 %0, %1, off"
                 :: "v"(lds_addr), "v"((unsigned long long)(uintptr_t)gptr)
                 : "memory");
}
__device__ __forceinline__ void async_wait_all() {
    asm volatile("s_wait_asynccnt 0" ::: "memory");
}

// ---------------- degree / norm ----------------

__global__ void k_init_deg(float* deg, int n) {
    int i = blockIdx.x * blockDim.x + threadIdx.x;
    if (i < n) deg[i] = 1.0f;  // self loop contributes 1
}

__global__ void k_accum_deg(const int* __restrict__ row, float* deg, int E) {
    int e = blockIdx.x * blockDim.x + threadIdx.x;
    if (e < E) unsafeAtomicAdd(&deg[row[e]], 1.0f);
}

__global__ void k_rsqrt_deg(float* deg, int n) {
    int i = blockIdx.x * blockDim.x + threadIdx.x;
    if (i < n) {
        float d = deg[i];
        deg[i] = (d > 0.0f) ? rsqrtf(d) : 0.0f;
    }
}

__global__ void k_edge_norm(const int* __restrict__ row, const int* __restrict__ col,
                            const float* __restrict__ dinv, float* __restrict__ nrm, int E) {
    int e = blockIdx.x * blockDim.x + threadIdx.x;
    if (e < E) nrm[e] = dinv[row[e]] * dinv[col[e]];
}

// ---------------- GEMM1: h = relu(x @ W1^T + b1); also hidden = h ----------------
// 8 waves/block; W1 (64KB) staged once per block into padded LDS.
__global__ void __launch_bounds__(256) k_gemm1_relu(
    const float* __restrict__ x, const float* __restrict__ W1,
    const float* __restrict__ b1, float* __restrict__ h,
    float* __restrict__ hidden, int M) {
    __shared__ float sW[DIM * PAD1];

    // stage W1[128][128] -> LDS (async, 16 float4 per thread)
    for (int i = threadIdx.x; i < DIM * (DIM / 4); i += 256) {
        int r = i >> 5;            // 32 float4 per row
        int c = (i & 31) << 2;
        async_lds_b128((unsigned)(uintptr_t)&sW[r * PAD1 + c], W1 + r * DIM + c);
    }
    async_wait_all();
    __syncthreads();

    int wave = (blockIdx.x * blockDim.x + threadIdx.x) >> 5;
    int lane = threadIdx.x & 31;
    if (wave < (M >> 4)) {                 // wave-uniform guard (EXEC all-ones)
        int m0 = wave << 4;
        int half  = lane >> 4;             // selects K pair {0,1} vs {2,3}
        int nlane = lane & 15;             // A: M row; B/D: N column
        const float* xrow = x + (size_t)(m0 + nlane) * DIM;

        v8f acc[8] = {};
        for (int k0 = 0; k0 < DIM; k0 += 4) {
            int ka = k0 + 2 * half;
            v2f a = *(const v2f*)(xrow + ka);
            #pragma unroll
            for (int t = 0; t < 8; t++) {
                // B^T[k][n] = W1[n][k] from LDS, n = t*16 + nlane
                v2f b = *(const v2f*)(&sW[(t * 16 + nlane) * PAD1 + ka]);
                acc[t] = __builtin_amdgcn_wmma_f32_16x16x4_f32(
                    false, a, false, b, 0, acc[t], false, false);
            }
        }
        #pragma unroll
        for (int t = 0; t < 8; t++) {
            int n = t * 16 + nlane;
            float bias = b1[n];
            #pragma unroll
            for (int v = 0; v < 8; v++) {
                int m = m0 + v + 8 * half;       // D: M = vgpr + 8*(lane/16)
                float val = acc[t][v] + bias;
                val = val > 0.0f ? val : 0.0f;
                size_t off = (size_t)m * DIM + n;
                h[off] = val;
                hidden[off] = val;
            }
        }
    }
}

// ---------------- propagation: dst[col] += src[row] * norm (scatter) ----------------

__global__ void k_zero4(float4* p, long n4) {
    long i = (long)blockIdx.x * blockDim.x + threadIdx.x;
    if (i < n4) p[i] = make_float4(0.f, 0.f, 0.f, 0.f);
}

__global__ void k_prop4(const int* __restrict__ row, const int* __restrict__ col,
                        const float* __restrict__ nrm, const float* __restrict__ dinv,
                        const float4* __restrict__ src, float* __restrict__ dst,
                        int E, int N) {
    long gid = (long)blockIdx.x * blockDim.x + threadIdx.x;
    long total = (long)(E + N) * (DIM / 4);
    if (gid >= total) return;
    int e  = (int)(gid >> 5);          // 32 float4 per feature row
    int d4 = (int)(gid & 31);
    int s, t;
    float w;
    if (e < E) {
        s = row[e]; t = col[e]; w = nrm[e];
    } else {
        s = t = e - E;                 // self loop
        float dv = dinv[s];
        w = dv * dv;
    }
    float4 v = src[(long)s * (DIM / 4) + d4];   // global_load_b128 gather
    float* dp = dst + (long)t * DIM + (d4 << 2);
    unsafeAtomicAdd(dp + 0, v.x * w);
    unsafeAtomicAdd(dp + 1, v.y * w);
    unsafeAtomicAdd(dp + 2, v.z * w);
    unsafeAtomicAdd(dp + 3, v.w * w);
}

__global__ void k_add4(float4* __restrict__ hidden, const float4* __restrict__ h, long n4) {
    long i = (long)blockIdx.x * blockDim.x + threadIdx.x;
    if (i < n4) {
        float4 a = hidden[i], b = h[i];
        hidden[i] = make_float4(a.x + b.x, a.y + b.y, a.z + b.z, a.w + b.w);
    }
}

// ---------------- L2 normalize rows (one wave per node, float4 per lane) ----------------

__global__ void k_l2norm(float* hidden, int n) {
    int wave = (blockIdx.x * blockDim.x + threadIdx.x) >> 5;
    int lane = threadIdx.x & 31;
    if (wave >= n) return;
    float4* p = (float4*)(hidden + (size_t)wave * DIM) + lane;
    float4 v = *p;
    float ss = v.x * v.x + v.y * v.y + v.z * v.z + v.w * v.w;
    #pragma unroll
    for (int m = 16; m >= 1; m >>= 1) ss += __shfl_xor(ss, m, 32);
    float scale = 1.0f / fmaxf(sqrtf(ss), 1e-12f);
    *p = make_float4(v.x * scale, v.y * scale, v.z * scale, v.w * scale);
}

// ---------------- head: relu(cat(h_i,h_j) @ Wl1^T + bl1) @ Wl2^T + bl2 ----------------
// W_l1 (128KB) staged once per block into padded LDS; 16x128 relu tile stays in
// registers; final 128-dot reduced with wave32 shfl_xor (no big intermediate).
__global__ void __launch_bounds__(256) k_head(
    const float* __restrict__ hidden, const float* __restrict__ Wl1,
    const float* __restrict__ bl1, const float* __restrict__ Wl2,
    const float* __restrict__ bl2, const int* __restrict__ ei,
    float* __restrict__ out, int L) {
    __shared__ float sW[DIM * PAD2];

    // stage Wl1[128][256] -> LDS (async, 32 float4 per thread)
    for (int i = threadIdx.x; i < DIM * (256 / 4); i += 256) {
        int r = i >> 6;            // 64 float4 per row
        int c = (i & 63) << 2;
        async_lds_b128((unsigned)(uintptr_t)&sW[r * PAD2 + c], Wl1 + r * 256 + c);
    }
    async_wait_all();
    __syncthreads();

    int wave = (blockIdx.x * blockDim.x + threadIdx.x) >> 5;
    int lane = threadIdx.x & 31;
    if (wave >= (L >> 4)) return;   // after barrier; wave-uniform
    int m0 = wave << 4;
    int half  = lane >> 4;
    int nlane = lane & 15;
    int mrow = m0 + nlane;          // A lane always holds row M = lane&15
    const float* pa0 = hidden + (size_t)ei[mrow] * DIM;      // k in [0,128)
    const float* pa1 = hidden + (size_t)ei[L + mrow] * DIM;  // k in [128,256)

    v8f acc[8] = {};
    for (int kh = 0; kh < 2; kh++) {
        const float* pa = kh ? pa1 : pa0;
        for (int k0 = 0; k0 < DIM; k0 += 4) {
            int ka = k0 + 2 * half;
            v2f a = *(const v2f*)(pa + ka);
            int kg = kh * DIM + ka;
            #pragma unroll
            for (int t = 0; t < 8; t++) {
                v2f b = *(const v2f*)(&sW[(t * 16 + nlane) * PAD2 + kg]);
                acc[t] = __builtin_amdgcn_wmma_f32_16x16x4_f32(
                    false, a, false, b, 0, acc[t], false, false);
            }
        }
    }
    // bias + relu + dot with Wl2 (each lane owns columns n = t*16 + nlane)
    float part[8];
    #pragma unroll
    for (int v = 0; v < 8; v++) part[v] = 0.0f;
    #pragma unroll
    for (int t = 0; t < 8; t++) {
        int n = t * 16 + nlane;
        float bias = bl1[n];
        float w2 = Wl2[n];
        #pragma unroll
        for (int v = 0; v < 8; v++) {
            float val = acc[t][v] + bias;
            val = val > 0.0f ? val : 0.0f;
            part[v] += val * w2;
        }
    }
    #pragma unroll
    for (int m = 8; m >= 1; m >>= 1) {
        #pragma unroll
        for (int v = 0; v < 8; v++) part[v] += __shfl_xor(part[v], m, 32);
    }
    if (nlane == 0) {
        float b2 = bl2[0];
        #pragma unroll
        for (int v = 0; v < 8; v++) out[m0 + v + 8 * half] = part[v] + b2;
    }
}

// ---------------- launcher ----------------

static inline int cdiv(long a, int b) { return (int)((a + b - 1) / b); }

extern "C" void kernel_launch(void* const* d_in, const int* in_sizes, int n_in,
                              void* d_out, int out_size, void* d_ws, size_t ws_size,
                              hipStream_t stream) {
    (void)n_in; (void)out_size; (void)ws_size;
    const float* x   = (const float*)d_in[0];
    const float* W1  = (const float*)d_in[1];
    const float* b1  = (const float*)d_in[2];
    const float* Wl1 = (const float*)d_in[3];
    const float* bl1 = (const float*)d_in[4];
    const float* Wl2 = (const float*)d_in[5];
    const float* bl2 = (const float*)d_in[6];
    const int*  eidx = (const int*)d_in[7];
    const int* elidx = (const int*)d_in[8];

    int N = in_sizes[0] / DIM;   // 50000 nodes
    int E = in_sizes[7] / 2;     // 500000 edges
    int L = in_sizes[8] / 2;     // 100000 label pairs

    const int* row = eidx;
    const int* col = eidx + E;

    // workspace layout (floats): dinv | norm | hA | hB | hidden  (~79 MB)
    float* ws     = (float*)d_ws;
    float* dinv   = ws;
    float* nrm    = dinv + N;
    float* hA     = nrm + E;
    float* hB     = hA + (size_t)N * DIM;
    float* hidden = hB + (size_t)N * DIM;
    long   nd4    = (long)N * (DIM / 4);

    const int B = 256;

    // 1) degree -> dinv -> per-edge norm
    k_init_deg  <<<cdiv(N, B), B, 0, stream>>>(dinv, N);
    k_accum_deg <<<cdiv(E, B), B, 0, stream>>>(row, dinv, E);
    k_rsqrt_deg <<<cdiv(N, B), B, 0, stream>>>(dinv, N);
    k_edge_norm <<<cdiv(E, B), B, 0, stream>>>(row, col, dinv, nrm, E);

    // 2) h = relu(x @ W1^T + b1), hidden = h   (WMMA f32, W1 in LDS)
    k_gemm1_relu<<<cdiv((N >> 4), 8), B, 0, stream>>>(x, W1, b1, hA, hidden, N);

    // 3) hop 1: hB = scatter(hA); hidden += hB
    long ptotal = (long)(E + N) * (DIM / 4);
    k_zero4<<<cdiv(nd4, B), B, 0, stream>>>((float4*)hB, nd4);
    k_prop4<<<cdiv(ptotal, B), B, 0, stream>>>(row, col, nrm, dinv, (const float4*)hA, hB, E, N);
    k_add4 <<<cdiv(nd4, B), B, 0, stream>>>((float4*)hidden, (const float4*)hB, nd4);

    // 4) hop 2: hA = scatter(hB); hidden += hA
    k_zero4<<<cdiv(nd4, B), B, 0, stream>>>((float4*)hA, nd4);
    k_prop4<<<cdiv(ptotal, B), B, 0, stream>>>(row, col, nrm, dinv, (const float4*)hB, hA, E, N);
    k_add4 <<<cdiv(nd4, B), B, 0, stream>>>((float4*)hidden, (const float4*)hA, nd4);

    // 5) L2 normalize rows
    k_l2norm<<<cdiv((long)N * 32, B), B, 0, stream>>>(hidden, N);

    // 6) fused link-prediction head -> d_out [L]  (Wl1 in LDS)
    k_head<<<cdiv((long)(L >> 4) * 32, B), B, 0, stream>>>(
        hidden, Wl1, bl1, Wl2, bl2, elidx, (float*)d_out, L);
}